// Block_mamba_8985071583556
// MI455X (gfx1250) — compile-verified
//
#include <hip/hip_runtime.h>
#include <math.h>

typedef __attribute__((ext_vector_type(2))) float v2f;
typedef __attribute__((ext_vector_type(8))) float v8f;

#define BB   8
#define LL   512
#define CC   192
#define DI   384
#define NSH  32       // SEG * BB
#define NS   48       // D_STATE
#define DTR  12       // DT_RANK
#define XD   108      // DT_RANK + 2*D_STATE
#define HD   384
#define TT   128      // LL / SEG
#define KMAX 384      // max GEMM K in this model
#define FEPS 1e-5f
#define TWOPI 6.283185307179586f

// ---------------------------------------------------------------------------
// Row layernorm helper: blockDim.x == CC (192). Returns (v-mean)*rstd.
// ---------------------------------------------------------------------------
__device__ __forceinline__ float rowln_norm(float v) {
  __shared__ float sv[CC];
  __shared__ float sp[64], sq[64];
  __shared__ float st[2];
  int c = threadIdx.x;
  sv[c] = v;
  __syncthreads();
  if (c < 64) {
    float a = 0.f, b = 0.f;
    for (int i = c; i < CC; i += 64) { float t = sv[i]; a += t; b += t * t; }
    sp[c] = a; sq[c] = b;
  }
  __syncthreads();
  if (c == 0) {
    float a = 0.f, b = 0.f;
    for (int i = 0; i < 64; ++i) { a += sp[i]; b += sq[i]; }
    float m = a / CC;
    st[0] = m;
    st[1] = rsqrtf(b / CC - m * m + FEPS);
  }
  __syncthreads();
  return (v - st[0]) * st[1];
}

// ---------------------------------------------------------------------------
// Build shifted copies xo and apply mamba layernorm -> xn (NSH, LL, CC)
// shift s: xo[s*BB+b, l] = x[b, l+s*TT] if l < LL - s*TT else x[b, l]
// ---------------------------------------------------------------------------
__global__ void k_prep(const float* __restrict__ x, const float* __restrict__ w,
                       const float* __restrict__ b, float* __restrict__ xn) {
  int row = blockIdx.x;                 // 0 .. NSH*LL-1
  int bb = row / LL, l = row % LL;
  int s = bb / BB, bi = bb % BB;
  int shift = s * TT;
  int lsrc = (l < LL - shift) ? (l + shift) : l;
  int c = threadIdx.x;
  float v = x[((size_t)bi * LL + lsrc) * CC + c];
  float r = rowln_norm(v);
  xn[(size_t)row * CC + c] = r * w[c] + b[c];
}

// ---------------------------------------------------------------------------
// WMMA fp32 GEMM: Out[m,n] = sum_k A[m,k]*W[n,k] (+bias[n]) (+softplus)
// V_WMMA_F32_16X16X4_F32. Block = 256 thr (8 waves) -> 32 rows x 128 cols.
// A strip (32 x K) staged in LDS once per block (shared by all 8 waves).
// Each wave computes TWO 16x16 tiles (rows 0-15 and 16-31 of the strip)
// sharing one B fragment -> 2 independent WMMA chains for ILP.
// Fragment element v lives at K = k + v + 2*(lane>=16).
// K multiple of 4, M multiple of 32 (true for all calls here).
// ---------------------------------------------------------------------------
__global__ void k_gemm_wmma(const float* __restrict__ A, const float* __restrict__ W,
                            const float* __restrict__ bias, float* __restrict__ Out,
                            int M, int N, int K, int lda, int ldw, int ldo, int act) {
  __shared__ float As[32 * KMAX];
  int wave = threadIdx.x >> 5;
  int lane = threadIdx.x & 31;
  int half = lane >> 4;
  int r = lane & 15;
  int mBase = blockIdx.y * 32;
  int nBase = (blockIdx.x * 8 + wave) * 16;

  // stage A strip into LDS: wave w stages rows 4w..4w+3 (coalesced)
  for (int rr = 0; rr < 4; ++rr) {
    int row = wave * 4 + rr;
    const float* src = A + (size_t)(mBase + row) * lda;
    for (int c = lane; c < K; c += 32) As[row * K + c] = src[c];
  }
  __syncthreads();

  if (nBase >= N) return;

  int n = nBase + r;
  int nc = (n < N) ? n : (N - 1);           // clamped valid address
  float nmask = (n < N) ? 1.f : 0.f;
  const float2* w2  = (const float2*)(W + (size_t)nc * ldw);
  const float2* a20 = (const float2*)(As + (size_t)r * K);
  const float2* a21 = (const float2*)(As + (size_t)(16 + r) * K);

  v8f acc0 = {}, acc1 = {};
  bool fullTile = (nBase + 16 <= N);        // wave-uniform
  if (fullTile) {
    for (int k = 0; k < K; k += 4) {
      int j = (k >> 1) + half;              // float2 index of element k+2*half
      float2 a0 = a20[j], a1 = a21[j], bw = w2[j];
      v2f af0 = {a0.x, a0.y}, af1 = {a1.x, a1.y};
      v2f bf  = {bw.x, bw.y};
      acc0 = __builtin_amdgcn_wmma_f32_16x16x4_f32(false, af0, false, bf,
                                                   (short)0, acc0, false, false);
      acc1 = __builtin_amdgcn_wmma_f32_16x16x4_f32(false, af1, false, bf,
                                                   (short)0, acc1, false, false);
    }
  } else {
    for (int k = 0; k < K; k += 4) {
      int j = (k >> 1) + half;
      float2 a0 = a20[j], a1 = a21[j], bw = w2[j];
      v2f af0 = {a0.x, a0.y}, af1 = {a1.x, a1.y};
      v2f bf  = {bw.x * nmask, bw.y * nmask};   // branchless zero-pad
      acc0 = __builtin_amdgcn_wmma_f32_16x16x4_f32(false, af0, false, bf,
                                                   (short)0, acc0, false, false);
      acc1 = __builtin_amdgcn_wmma_f32_16x16x4_f32(false, af1, false, bf,
                                                   (short)0, acc1, false, false);
    }
  }

  if (n < N) {
    float bv = bias ? bias[n] : 0.f;
#pragma unroll
    for (int v = 0; v < 8; ++v) {
      int mm0 = mBase + v + 8 * half;
      int mm1 = mm0 + 16;
      float o0 = acc0[v] + bv;
      float o1 = acc1[v] + bv;
      if (act == 1) {
        o0 = (o0 > 20.f) ? o0 : log1pf(__expf(o0));   // softplus
        o1 = (o1 > 20.f) ? o1 : log1pf(__expf(o1));
      }
      Out[(size_t)mm0 * ldo + n] = o0;
      Out[(size_t)mm1 * ldo + n] = o1;
    }
  }
}

// ---------------------------------------------------------------------------
// Depthwise causal conv (k=4) + SiLU.  xa = xz[..., :DI]  (row stride 2*DI)
// ---------------------------------------------------------------------------
__global__ void k_conv(const float* __restrict__ xz, const float* __restrict__ cw,
                       const float* __restrict__ cb, float* __restrict__ xc) {
  int idx = blockIdx.x * blockDim.x + threadIdx.x;
  const int total = NSH * LL * DI;
  if (idx >= total) return;
  int d = idx % DI;
  int l = (idx / DI) % LL;
  int bb = idx / (DI * LL);
  float acc = cb[d];
#pragma unroll
  for (int k = 0; k < 4; ++k) {
    int ls = l + k - 3;
    if (ls >= 0)
      acc += xz[((size_t)bb * LL + ls) * (2 * DI) + d] * cw[d * 4 + k];
  }
  xc[idx] = acc / (1.f + __expf(-acc));  // silu
}

// ---------------------------------------------------------------------------
// Selective scan.  One block per sequence (NSH), one lane per channel (DI).
// h[NS] state in registers, B/C vectors staged in LDS per timestep.
// Output gated:  y = (scan + u*D) * silu(z)
// ---------------------------------------------------------------------------
__global__ void k_scan(const float* __restrict__ xdbl, const float* __restrict__ dt,
                       const float* __restrict__ xc, const float* __restrict__ xz,
                       const float* __restrict__ A_log, const float* __restrict__ Dp,
                       float* __restrict__ yg) {
  int bb = blockIdx.x;
  int d = threadIdx.x;
  float Areg[NS], h[NS];
#pragma unroll
  for (int n = 0; n < NS; ++n) {
    Areg[n] = -__expf(A_log[d * NS + n]);
    h[n] = 0.f;
  }
  float Dv = Dp[d];
  __shared__ float sB[NS], sC[NS];
  for (int l = 0; l < LL; ++l) {
    const float* row = xdbl + ((size_t)bb * LL + l) * XD;
    if (d < 2 * NS) {
      float v = row[DTR + d];
      if (d < NS) sB[d] = v; else sC[d - NS] = v;
    }
    __syncthreads();
    size_t off = ((size_t)bb * LL + l) * DI + d;
    float dtv = dt[off];
    float u = xc[off];
    float du = dtv * u;
    float y = 0.f;
#pragma unroll
    for (int n = 0; n < NS; ++n) {
      float dA = __expf(dtv * Areg[n]);
      h[n] = dA * h[n] + du * sB[n];
      y += h[n] * sC[n];
    }
    y += u * Dv;
    float z = xz[((size_t)bb * LL + l) * (2 * DI) + DI + d];
    y *= z / (1.f + __expf(-z));
    yg[off] = y;
    __syncthreads();
  }
}

// ---------------------------------------------------------------------------
// Segment recombination:  xo0[b,l] = (ys[0][b,l] + sum_j ys[j+1][...]) / (i+1)
// ---------------------------------------------------------------------------
__global__ void k_recomb(const float* __restrict__ yp, float* __restrict__ xo0) {
  int idx = blockIdx.x * blockDim.x + threadIdx.x;
  const int total = BB * LL * CC;
  if (idx >= total) return;
  int c = idx % CC;
  int l = (idx / CC) % LL;
  int b = idx / (CC * LL);
  int i = l / TT;
  int lloc = l - i * TT;
  float s = yp[(((size_t)0 * BB + b) * LL + l) * CC + c];
  for (int j = 0; j < i; ++j) {
    int ls = TT * (i - j - 1) + lloc;
    s += yp[(((size_t)(j + 1) * BB + b) * LL + ls) * CC + c];
  }
  xo0[idx] = s / (float)(i + 1);
}

// ---------------------------------------------------------------------------
// Generic row layernorm over CC, with optional residual added AFTER the LN.
// ---------------------------------------------------------------------------
__global__ void k_rowln(const float* __restrict__ in, const float* __restrict__ res,
                        const float* __restrict__ w, const float* __restrict__ b,
                        float* __restrict__ out) {
  int row = blockIdx.x;
  int c = threadIdx.x;
  size_t off = (size_t)row * CC + c;
  float v = in[off];
  float o = rowln_norm(v) * w[c] + b[c];
  if (res) o += res[off];
  out[off] = o;
}

// ---------------------------------------------------------------------------
// Batchnorm statistics per channel over (B*L) rows.
// ---------------------------------------------------------------------------
__global__ void k_bnstat(const float* __restrict__ in, int rows, int Cch,
                         float* __restrict__ mean, float* __restrict__ rstd) {
  int ch = blockIdx.x;
  float s = 0.f, q = 0.f;
  for (int r = threadIdx.x; r < rows; r += blockDim.x) {
    float v = in[(size_t)r * Cch + ch];
    s += v; q += v * v;
  }
  __shared__ float ss[256], sq[256];
  ss[threadIdx.x] = s; sq[threadIdx.x] = q;
  __syncthreads();
  for (int st = 128; st > 0; st >>= 1) {
    if (threadIdx.x < st) {
      ss[threadIdx.x] += ss[threadIdx.x + st];
      sq[threadIdx.x] += sq[threadIdx.x + st];
    }
    __syncthreads();
  }
  if (threadIdx.x == 0) {
    float m = ss[0] / rows;
    float var = sq[0] / rows - m * m;
    mean[ch] = m;
    rstd[ch] = rsqrtf(var + FEPS);
  }
}

// ---------------------------------------------------------------------------
// Radix-2 Stockham FFT (length 512) in LDS. 256 threads = 1 butterfly each.
// ---------------------------------------------------------------------------
__device__ __forceinline__ void fft512(float (*re)[LL], float (*im)[LL],
                                       int& cur, float sgn) {
  for (int n = LL, s = 1; n > 1; n >>= 1, s <<= 1) {
    int mhalf = n >> 1;
    int t = threadIdx.x;                // 256 butterflies per stage
    int p = t / s;
    int q = t - p * s;
    float ang = sgn * TWOPI * (float)p / (float)n;
    float wr = __cosf(ang), wi = __sinf(ang);
    float ar = re[cur][q + s * p],            ai = im[cur][q + s * p];
    float br = re[cur][q + s * (p + mhalf)],  bi = im[cur][q + s * (p + mhalf)];
    int nxt = cur ^ 1;
    re[nxt][q + s * (2 * p)]     = ar + br;
    im[nxt][q + s * (2 * p)]     = ai + bi;
    float dr = ar - br, di = ai - bi;
    re[nxt][q + s * (2 * p + 1)] = dr * wr - di * wi;
    im[nxt][q + s * (2 * p + 1)] = dr * wi + di * wr;
    __syncthreads();
    cur = nxt;
  }
}

// ---------------------------------------------------------------------------
// Fused: BN1+ReLU -> FFT(ortho) -> diag complex mix + ReLU -> IFFT(ortho).real
// One block per (batch, channel). blockDim.x must be 256.
// ---------------------------------------------------------------------------
__global__ void k_fftmix(const float* __restrict__ h, const float* __restrict__ m1,
                         const float* __restrict__ r1, const float* __restrict__ g1,
                         const float* __restrict__ b1, const float* __restrict__ rw,
                         const float* __restrict__ iw, const float* __restrict__ rb,
                         const float* __restrict__ ib, float* __restrict__ xt) {
  int ch = blockIdx.x % HD;
  int b = blockIdx.x / HD;
  __shared__ float re[2][LL], im[2][LL];
  for (int l = threadIdx.x; l < LL; l += blockDim.x) {
    float v = h[((size_t)b * LL + l) * HD + ch];
    v = (v - m1[ch]) * r1[ch] * g1[ch] + b1[ch];
    re[0][l] = fmaxf(v, 0.f);
    im[0][l] = 0.f;
  }
  __syncthreads();
  int cur = 0;
  fft512(re, im, cur, -1.f);               // forward
  const float scale = rsqrtf((float)LL);   // ortho norm each direction
  float rd = rw[(size_t)ch * HD + ch];
  float ig = iw[(size_t)ch * HD + ch];
  for (int l = threadIdx.x; l < LL; l += blockDim.x) {
    float a = re[cur][l] * scale;
    float c = im[cur][l] * scale;
    re[cur][l] = fmaxf(a * rd - c * ig + rb[ch], 0.f);
    im[cur][l] = fmaxf(c * rd + a * ig + ib[ch], 0.f);
  }
  __syncthreads();
  fft512(re, im, cur, 1.f);                // inverse
  for (int l = threadIdx.x; l < LL; l += blockDim.x)
    xt[((size_t)b * LL + l) * HD + ch] = re[cur][l] * scale;
}

// ---------------------------------------------------------------------------
// Final: out = x1 + BN2(fc2_out)
// ---------------------------------------------------------------------------
__global__ void k_final(const float* __restrict__ x1, const float* __restrict__ f2,
                        const float* __restrict__ m2, const float* __restrict__ r2,
                        const float* __restrict__ g2, const float* __restrict__ b2,
                        float* __restrict__ out) {
  int idx = blockIdx.x * blockDim.x + threadIdx.x;
  const int total = BB * LL * CC;
  if (idx >= total) return;
  int c = idx % CC;
  out[idx] = x1[idx] + (f2[idx] - m2[c]) * r2[c] * g2[c] + b2[c];
}

// ---------------------------------------------------------------------------
extern "C" void kernel_launch(void* const* d_in, const int* in_sizes, int n_in,
                              void* d_out, int out_size, void* d_ws, size_t ws_size,
                              hipStream_t stream) {
  (void)in_sizes; (void)n_in; (void)out_size; (void)ws_size;
  const float* x         = (const float*)d_in[0];
  const float* in_proj_w = (const float*)d_in[1];
  const float* conv_w    = (const float*)d_in[2];
  const float* conv_b    = (const float*)d_in[3];
  const float* x_proj_w  = (const float*)d_in[4];
  const float* dt_proj_w = (const float*)d_in[5];
  const float* dt_proj_b = (const float*)d_in[6];
  const float* A_log     = (const float*)d_in[7];
  const float* D_param   = (const float*)d_in[8];
  const float* out_proj_w= (const float*)d_in[9];
  const float* mnorm_w   = (const float*)d_in[10];
  const float* mnorm_b   = (const float*)d_in[11];
  const float* norm1_w   = (const float*)d_in[12];
  const float* norm1_b   = (const float*)d_in[13];
  const float* norm2_w   = (const float*)d_in[14];
  const float* norm2_b   = (const float*)d_in[15];
  const float* fc1_w     = (const float*)d_in[16];
  const float* bn1_g     = (const float*)d_in[17];
  const float* bn1_b     = (const float*)d_in[18];
  const float* r_w       = (const float*)d_in[19];
  const float* i_w       = (const float*)d_in[20];
  const float* rb        = (const float*)d_in[21];
  const float* ib        = (const float*)d_in[22];
  const float* fc2_w     = (const float*)d_in[23];
  const float* bn2_g     = (const float*)d_in[24];
  const float* bn2_b     = (const float*)d_in[25];

  // workspace layout
  float* p = (float*)d_ws;
  float* xn   = p; p += (size_t)NSH * LL * CC;
  float* xz   = p; p += (size_t)NSH * LL * 2 * DI;
  float* xc   = p; p += (size_t)NSH * LL * DI;
  float* xdbl = p; p += (size_t)NSH * LL * XD;
  float* dt   = p; p += (size_t)NSH * LL * DI;
  float* yg   = p; p += (size_t)NSH * LL * DI;
  float* yp   = p; p += (size_t)NSH * LL * CC;
  float* xo0  = p; p += (size_t)BB * LL * CC;
  float* x1   = p; p += (size_t)BB * LL * CC;
  float* t2   = p; p += (size_t)BB * LL * CC;
  float* hb   = p; p += (size_t)BB * LL * HD;
  float* xt   = p; p += (size_t)BB * LL * HD;
  float* f2   = p; p += (size_t)BB * LL * CC;
  float* m1   = p; p += HD;
  float* r1   = p; p += HD;
  float* m2   = p; p += CC;
  float* r2   = p; p += CC;

  const int Mrows = NSH * LL;   // 16384
  const int Frows = BB * LL;    // 4096

  // 1) shifts + mamba layernorm
  k_prep<<<Mrows, CC, 0, stream>>>(x, mnorm_w, mnorm_b, xn);
  // 2) in_proj: (16384,192)x(768,192)^T
  k_gemm_wmma<<<dim3((2 * DI + 127) / 128, Mrows / 32), 256, 0, stream>>>(
      xn, in_proj_w, nullptr, xz, Mrows, 2 * DI, CC, CC, CC, 2 * DI, 0);
  // 3) depthwise conv + silu
  k_conv<<<(NSH * LL * DI + 255) / 256, 256, 0, stream>>>(xz, conv_w, conv_b, xc);
  // 4) x_proj: (16384,384)x(108,384)^T
  k_gemm_wmma<<<dim3((XD + 127) / 128, Mrows / 32), 256, 0, stream>>>(
      xc, x_proj_w, nullptr, xdbl, Mrows, XD, DI, DI, DI, XD, 0);
  // 5) dt_proj + softplus: (16384,12)x(384,12)^T  (A = dtl part of xdbl)
  k_gemm_wmma<<<dim3((DI + 127) / 128, Mrows / 32), 256, 0, stream>>>(
      xdbl, dt_proj_w, dt_proj_b, dt, Mrows, DI, DTR, XD, DTR, DI, 1);
  // 6) selective scan + D skip + gating
  k_scan<<<NSH, DI, 0, stream>>>(xdbl, dt, xc, xz, A_log, D_param, yg);
  // 7) out_proj: (16384,384)x(192,384)^T
  k_gemm_wmma<<<dim3((CC + 127) / 128, Mrows / 32), 256, 0, stream>>>(
      yg, out_proj_w, nullptr, yp, Mrows, CC, DI, DI, DI, CC, 0);
  // 8) segment recombination, then x1 = x + LN1(xo0)
  k_recomb<<<(Frows * CC + 255) / 256, 256, 0, stream>>>(yp, xo0);
  k_rowln<<<Frows, CC, 0, stream>>>(xo0, x, norm1_w, norm1_b, x1);
  // 9) FFN: t2 = LN2(x1); h = t2 @ fc1^T
  k_rowln<<<Frows, CC, 0, stream>>>(x1, nullptr, norm2_w, norm2_b, t2);
  k_gemm_wmma<<<dim3((HD + 127) / 128, Frows / 32), 256, 0, stream>>>(
      t2, fc1_w, nullptr, hb, Frows, HD, CC, CC, CC, HD, 0);
  // 10) BN1 stats; fused BN1+ReLU -> FFT -> mix -> IFFT
  k_bnstat<<<HD, 256, 0, stream>>>(hb, Frows, HD, m1, r1);
  k_fftmix<<<BB * HD, 256, 0, stream>>>(hb, m1, r1, bn1_g, bn1_b,
                                        r_w, i_w, rb, ib, xt);
  // 11) fc2, BN2 stats, final residual
  k_gemm_wmma<<<dim3((CC + 127) / 128, Frows / 32), 256, 0, stream>>>(
      xt, fc2_w, nullptr, f2, Frows, CC, HD, HD, HD, CC, 0);
  k_bnstat<<<CC, 256, 0, stream>>>(f2, Frows, CC, m2, r2);
  k_final<<<(Frows * CC + 255) / 256, 256, 0, stream>>>(
      x1, f2, m2, r2, bn2_g, bn2_b, (float*)d_out);
}